// DeformableSentenceSplit_33423435497598
// MI455X (gfx1250) — compile-verified
//
#include <hip/hip_runtime.h>

// Problem constants (from reference)
#define BATCH 16
#define TLEN  2048
#define EMB   768
#define NSENT 32
#define SLEN  64
#define NOUT  64          // 2*NSENT
#define NCHUNK 16         // t-chunks for pooling
#define CHUNK_T 128       // TLEN / NCHUNK

typedef __attribute__((ext_vector_type(2))) float v2f;
typedef __attribute__((ext_vector_type(8))) float v8f;

// ---------------------------------------------------------------------------
// Phase 1a: partial sums over T. grid = BATCH*NCHUNK blocks, 256 threads.
// Each block handles 128 rows of one batch; each thread owns 3 embedding lanes.
// Fully coalesced: 256 consecutive f32 per load wavefront.
// ---------------------------------------------------------------------------
__global__ void pool_partial_kernel(const float* __restrict__ in,
                                    float* __restrict__ part) {
  const int blk = blockIdx.x;            // b*NCHUNK + c
  const int b   = blk >> 4;
  const int c   = blk & 15;
  const int tid = threadIdx.x;           // 0..255
  const float* base = in + ((size_t)b * TLEN + (size_t)c * CHUNK_T) * EMB;
  float s0 = 0.f, s1 = 0.f, s2 = 0.f;
  for (int t = 0; t < CHUNK_T; ++t) {
    const float* row = base + (size_t)t * EMB;
    s0 += row[tid];
    s1 += row[tid + 256];
    s2 += row[tid + 512];
  }
  float* p = part + (size_t)blk * EMB;
  p[tid]       = s0;
  p[tid + 256] = s1;
  p[tid + 512] = s2;
}

// ---------------------------------------------------------------------------
// Phase 1b: deterministic reduction of the 16 chunk partials -> pooled mean.
// grid = BATCH*EMB/256 = 48 blocks.
// ---------------------------------------------------------------------------
__global__ void pool_reduce_kernel(const float* __restrict__ part,
                                   float* __restrict__ pooled) {
  const int idx = blockIdx.x * 256 + threadIdx.x;   // 0 .. BATCH*EMB-1
  const int b = idx / EMB;
  const int e = idx - b * EMB;
  float s = 0.f;
#pragma unroll
  for (int c = 0; c < NCHUNK; ++c)
    s += part[((size_t)(b * NCHUNK + c)) * EMB + e];
  pooled[idx] = s * (1.0f / (float)TLEN);
}

// ---------------------------------------------------------------------------
// Phase 2: offsets = pooled @ W + b via V_WMMA_F32_16X16X4_F32, then index math.
// One block, 128 threads = 4 wave32s; wave w computes N-tile [16w, 16w+16).
// A (16x4 f32): lane L holds M=L%16, K = k0 + 2*(L>=16) + {0,1}  (2 VGPRs)
// B (4x16 f32): lane L holds N=L%16, same K split                (2 VGPRs)
// D (16x16 f32): VGPR r, lane L -> M = r + 8*(L>=16), N = L%16   (8 VGPRs)
// ---------------------------------------------------------------------------
__global__ void dense_index_kernel(const float* __restrict__ pooled,
                                   const float* __restrict__ W,
                                   const float* __restrict__ bias,
                                   int* __restrict__ sidx,
                                   int* __restrict__ slen) {
  __shared__ float offs_s[BATCH * NOUT];

  const int tid  = threadIdx.x;       // 0..127
  const int lane = tid & 31;
  const int wave = tid >> 5;          // N tile id
  const int half = lane >> 4;         // K sub-split
  const int m    = lane & 15;         // A: batch row
  const int n    = wave * 16 + (lane & 15);  // B/D: output column

  v8f acc = {};
  const float* arow = pooled + (size_t)m * EMB;
  for (int k0 = 0; k0 < EMB; k0 += 4) {
    const int kb = k0 + (half << 1);
    v2f a;
    a.x = arow[kb];
    a.y = arow[kb + 1];
    v2f bm;
    bm.x = W[(size_t)kb * NOUT + n];
    bm.y = W[(size_t)(kb + 1) * NOUT + n];
    acc = __builtin_amdgcn_wmma_f32_16x16x4_f32(
        /*neg_a=*/false, a, /*neg_b=*/false, bm,
        /*c_mod=*/(short)0, acc, /*reuse_a=*/false, /*reuse_b=*/false);
  }

  const float bv = bias[n];
#pragma unroll
  for (int r = 0; r < 8; ++r) {
    const int mm = r + half * 8;
    offs_s[mm * NOUT + n] = acc[r] + bv;
  }
  __syncthreads();

  // 512 (b,s) pairs -> start index + valid length, matching the reference's
  // clip(.,0,L-1).astype(int32) (truncation) and the two index clamps.
  for (int i = tid; i < BATCH * NSENT; i += 128) {
    const int bb = i >> 5;
    const int s  = i & 31;
    const float so_f = offs_s[bb * NOUT + s];
    const float eo_f = offs_s[bb * NOUT + NSENT + s];
    const int so = (int)fminf(fmaxf(so_f, 0.0f), (float)(SLEN - 1));
    const int eo = (int)fminf(fmaxf(eo_f, 0.0f), (float)(SLEN - 1));
    const int base  = s * SLEN;
    int start = base + so;
    if (start > TLEN - SLEN) start = TLEN - SLEN;
    int end = base + SLEN + eo;
    if (end > TLEN) end = TLEN;
    if (end < start) end = start;
    sidx[i] = start;
    slen[i] = end - start;
  }
}

// ---------------------------------------------------------------------------
// Phase 3: gather + zero-pad. One block per output row (b,s,j); 192 threads
// move 192 float4 (= 768 f32). Block-uniform predicate -> no divergence.
// start <= TLEN-SLEN guarantees start+j <= TLEN-1, so reads never go OOB.
// ---------------------------------------------------------------------------
__global__ void gather_kernel(const float* __restrict__ in,
                              const int* __restrict__ sidx,
                              const int* __restrict__ slen,
                              float* __restrict__ out) {
  const int row = blockIdx.x;          // 0 .. BATCH*NSENT*SLEN-1
  const int t   = threadIdx.x;         // 0..191
  const int j   = row & (SLEN - 1);
  const int bs  = row >> 6;            // b*NSENT + s
  const int b   = bs >> 5;

  const int start = sidx[bs];
  const int len   = slen[bs];

  float4 v = make_float4(0.f, 0.f, 0.f, 0.f);
  if (j < len) {
    const float4* src =
        (const float4*)(in + ((size_t)b * TLEN + (size_t)(start + j)) * EMB);
    v = src[t];
  }
  float4* dst = (float4*)(out + (size_t)row * EMB);
  dst[t] = v;
}

// ---------------------------------------------------------------------------
extern "C" void kernel_launch(void* const* d_in, const int* in_sizes, int n_in,
                              void* d_out, int out_size, void* d_ws, size_t ws_size,
                              hipStream_t stream) {
  (void)in_sizes; (void)n_in; (void)out_size; (void)ws_size;

  const float* in   = (const float*)d_in[0];  // [B, T, E]
  const float* W    = (const float*)d_in[1];  // [E, 2S]
  const float* bias = (const float*)d_in[2];  // [2S]
  float* out = (float*)d_out;                 // [B, S, L, E]

  // Workspace layout (every region fully written before being read):
  float* part   = (float*)d_ws;                         // [B*NCHUNK][EMB]
  float* pooled = part + (size_t)BATCH * NCHUNK * EMB;  // [B][EMB]
  int*   sidx   = (int*)(pooled + (size_t)BATCH * EMB); // [B*S]
  int*   slen   = sidx + BATCH * NSENT;                 // [B*S]
  // total: (196608 + 12288)*4 + 1024*4 bytes ~= 0.84 MB

  pool_partial_kernel<<<BATCH * NCHUNK, 256, 0, stream>>>(in, part);
  pool_reduce_kernel<<<(BATCH * EMB) / 256, 256, 0, stream>>>(part, pooled);
  dense_index_kernel<<<1, 128, 0, stream>>>(pooled, W, bias, sidx, slen);
  gather_kernel<<<BATCH * NSENT * SLEN, 192, 0, stream>>>(in, sidx, slen, out);
}